// SimpleLSTM_7988639170819
// MI455X (gfx1250) — compile-verified
//
#include <hip/hip_runtime.h>
#include <math.h>

// LSTM: B=128, T=8192, H=96, input dim 1. One workgroup per 16-batch slice,
// 24 waves per workgroup (one 16-wide gate tile each), f32 WMMA 16x16x4.
// x*w_ih is fused into the WMMA as a 25th K-chunk (K=96 -> 100); the gate
// bias rides in as the initial C accumulator. Activations use hardware
// V_TANH_F32 trans ops (branchless).

#define B_TOT 128
#define T_LEN 8192
#define HDIM  96
#define MB    16                 // batch rows per workgroup (WMMA M tile)
#define NWAVE 24                 // 4H/16 = 24 gate tiles, one per wave
#define NTHREADS (NWAVE * 32)
#define HSTR  100                // h row: 96 h + x at col 96 + zeros 97..99
#define KCH   25                 // 25 K-chunks of 4 (96 h + 4 [x,0,0,0])

typedef __attribute__((ext_vector_type(2))) float v2f;
typedef __attribute__((ext_vector_type(8))) float v8f;

__device__ __forceinline__ float fast_tanh(float v) {
#if __has_builtin(__builtin_amdgcn_tanhf)
    return __builtin_amdgcn_tanhf(v);          // v_tanh_f32 (1 trans op)
#else
    // branchless: tanh(v) = 2/(1 + 2^(-2*log2(e)*v)) - 1  (v_exp/v_rcp)
    float e = __builtin_amdgcn_exp2f(-2.8853900817779268f * v);
    return __builtin_amdgcn_rcpf(1.0f + e) * 2.0f - 1.0f;
#endif
}

__global__ __launch_bounds__(NTHREADS)
void lstm_wmma_f32_kernel(const float* __restrict__ x,     // [B,T]
                          const float* __restrict__ w_ih,  // [384]
                          const float* __restrict__ w_hh,  // [384,96]
                          const float* __restrict__ b_ih,  // [384]
                          const float* __restrict__ b_hh,  // [384]
                          const float* __restrict__ w_out, // [96]
                          const float* __restrict__ b_out, // [1]
                          float* __restrict__ out)         // [B,T]
{
    __shared__ float h_lds[MB * HSTR];        // [h | x | 0 0 0] per row
    __shared__ float gates[4 * MB * HDIM];    // activated i,f,g,o
    __shared__ float xbuf[2][MB];             // double-buffered x prefetch
    __shared__ float oacc[2][MB];             // double-buffered out-dot accum

    const int tid  = threadIdx.x;
    const int wid  = tid >> 5;                // 0..23 : gate N-tile
    const int lane = tid & 31;
    const int b0   = blockIdx.x * MB;

    // ---- per-lane constants for the activation phase (C/D tile layout) ----
    const int ncol = lane & 15;               // N within tile
    const int mhi  = (lane >> 4) << 3;        // rows r or r+8
    const int gcol = wid * 16 + ncol;         // global gate index 0..383
    const int gi   = wid / 6;                 // 0=i 1=f 2=g 3=o
    const int gofs = gi * (MB * HDIM) + (wid % 6) * 16 + ncol;
    // branchless activation: a = s*tanh(s*v) + o  (tanh for g, sigmoid else)
    const float act_s = (gi == 2) ? 1.0f : 0.5f;
    const float act_o = (gi == 2) ? 0.0f : 0.5f;

    // loop-invariant bias broadcast as initial WMMA C accumulator
    const float bias_c = b_ih[gcol] + b_hh[gcol];
    v8f biasC;
#pragma unroll
    for (int r = 0; r < 8; ++r) biasC[r] = bias_c;

    // ---- [w_hh^T | w_ih] B-fragments in registers, loaded once ----
    // B 4x16 f32 layout: VGPR0 = K=base, VGPR1 = K=base+1, base = 4kc + 2*(lane/16)
    const int kbase = (lane >> 4) << 1;
    v2f bfrag[KCH];
    {
        const float* wrow = w_hh + gcol * HDIM;   // row 'gcol' of w_hh
#pragma unroll
        for (int kc = 0; kc < 24; ++kc)
            bfrag[kc] = *(const v2f*)(wrow + 4 * kc + kbase);
        // chunk 24: K=96..99 -> [w_ih[gcol], 0] for lanes<16, [0,0] for >=16
        v2f bx = {0.f, 0.f};
        if (lane < 16) bx.x = w_ih[gcol];
        bfrag[24] = bx;
    }

    // ---- update-phase ownership: 2 consecutive state elems per thread ----
    const int e0 = tid * 2;                   // 768*2 = 1536 = 16*96
    const int ub = e0 / HDIM;
    const int uj = e0 % HDIM;                 // e0 even, never crosses a row
    const float wo0 = w_out[uj], wo1 = w_out[uj + 1];
    float c0 = 0.f, c1 = 0.f;
    const float bout = b_out[0];

    // ---- init shared state ----
    for (int i = tid; i < MB * HSTR; i += NTHREADS) h_lds[i] = 0.f;
    if (tid < MB) {
        oacc[0][tid] = 0.f;
        oacc[1][tid] = 0.f;
        xbuf[1][tid] = x[(b0 + tid) * T_LEN + 1];
    }
    __syncthreads();
    if (tid < MB)  // after zero-fill: x for t=0 into h-row column 96
        h_lds[tid * HSTR + 96] = x[(b0 + tid) * T_LEN + 0];
    __syncthreads();

    for (int t = 0; t < T_LEN; ++t) {
        const int p = t & 1;

        // ===== phase A: gates = [h|x] @ [w_hh|w_ih]^T  (f32 WMMA, K=100) ===
        // A 16x4 f32 layout: lane<16 holds (K=4kc, 4kc+1) of row lane,
        //                    lane>=16 holds (K=4kc+2, 4kc+3) of row lane-16.
        v8f acc0 = biasC;                     // bias rides as initial C
        v8f acc1 = {};                        // 2 chains break WMMA RAW dep
        {
            const float* hrow = h_lds + (lane & 15) * HSTR + kbase;
#pragma unroll
            for (int kc = 0; kc < KCH; ++kc) {
                v2f afrag = *(const v2f*)(hrow + 4 * kc);
                if (kc & 1)
                    acc1 = __builtin_amdgcn_wmma_f32_16x16x4_f32(
                               false, afrag, false, bfrag[kc],
                               (short)0, acc1, false, false);
                else
                    acc0 = __builtin_amdgcn_wmma_f32_16x16x4_f32(
                               false, afrag, false, bfrag[kc],
                               (short)0, acc0, false, false);
            }
        }
        // branchless activation, stage gate tile to LDS
#pragma unroll
        for (int r = 0; r < 8; ++r) {
            const int m = r + mhi;
            float v = acc0[r] + acc1[r];
            gates[gofs + m * HDIM] = act_s * fast_tanh(act_s * v) + act_o;
        }
        __syncthreads();

        // ===== phase B: c/h update + fused output dot + x(t+1) staging =====
        {
            const int base = ub * HDIM + uj;
            v2f iv = *(const v2f*)&gates[0 * MB * HDIM + base];
            v2f fv = *(const v2f*)&gates[1 * MB * HDIM + base];
            v2f gv = *(const v2f*)&gates[2 * MB * HDIM + base];
            v2f ov = *(const v2f*)&gates[3 * MB * HDIM + base];
            c0 = fv.x * c0 + iv.x * gv.x;
            c1 = fv.y * c1 + iv.y * gv.y;
            float h0 = ov.x * fast_tanh(c0);
            float h1 = ov.y * fast_tanh(c1);
            h_lds[ub * HSTR + uj]     = h0;
            h_lds[ub * HSTR + uj + 1] = h1;
            atomicAdd(&oacc[p][ub], h0 * wo0 + h1 * wo1);   // ds_add_f32
            if (tid < MB)  // x for step t+1 into h-row column 96
                h_lds[tid * HSTR + 96] = xbuf[(t + 1) & 1][tid];
        }
        __syncthreads();

        // ===== phase C (overlaps next phase A): drain + prefetch ===========
        if (tid < MB) {
            out[(b0 + tid) * T_LEN + t] = oacc[p][tid] + bout;
            oacc[p][tid] = 0.f;                     // reused at t+2
            if (t + 2 < T_LEN)
                xbuf[p][tid] = x[(b0 + tid) * T_LEN + t + 2];  // for t+2
        }
    }
}

extern "C" void kernel_launch(void* const* d_in, const int* in_sizes, int n_in,
                              void* d_out, int out_size, void* d_ws, size_t ws_size,
                              hipStream_t stream) {
    (void)in_sizes; (void)n_in; (void)d_ws; (void)ws_size; (void)out_size;
    const float* x     = (const float*)d_in[0];
    const float* w_ih  = (const float*)d_in[1];
    const float* w_hh  = (const float*)d_in[2];
    const float* b_ih  = (const float*)d_in[3];
    const float* b_hh  = (const float*)d_in[4];
    const float* w_out = (const float*)d_in[5];
    const float* b_out = (const float*)d_in[6];
    float* out = (float*)d_out;

    dim3 grid(B_TOT / MB);      // 8 workgroups, one per 16-batch slice
    dim3 block(NTHREADS);       // 24 waves of 32
    lstm_wmma_f32_kernel<<<grid, block, 0, stream>>>(
        x, w_ih, w_hh, b_ih, b_hh, w_out, b_out, out);
}